// Despawn2D_39393440039487
// MI455X (gfx1250) — compile-verified
//
#include <hip/hip_runtime.h>

typedef __attribute__((ext_vector_type(2))) float v2f;
typedef __attribute__((ext_vector_type(8))) float v8f;

#define ROWS   2048
#define M_IN   3000
#define LPAD   4096
#define P0     548
#define LEVELS 8

// h_lo[k] = scaling[lvl][k]
__device__ __forceinline__ float hlo(const float* __restrict__ sc, int lvl, int k) {
    return sc[lvl * 8 + k];
}
// h_hi[k] = scaling_rec[lvl][7-k] * (-1)^k   (flip + alternate-sign, from _make_wavelet)
__device__ __forceinline__ float hhi(const float* __restrict__ sr, int lvl, int k) {
    float v = sr[lvl * 8 + (7 - k)];
    return (k & 1) ? -v : v;
}

__global__ __launch_bounds__(32)
void despawn2d_wavelet_kernel(const float* __restrict__ x,
                              const float* __restrict__ scal,
                              const float* __restrict__ screc,
                              float* __restrict__ out)
{
    __shared__ float sA[LPAD];      // ping approx buffer (max 4096)
    __shared__ float sB[LPAD / 2];  // pong approx buffer (max 2048)
    __shared__ float sC[LPAD];      // all wavelet coeffs for this row (4096)

    const int row  = blockIdx.x;
    if (row >= ROWS) return;
    const int lane = threadIdx.x;        // wave32: 0..31
    const int m    = lane & 15;          // M index for A / N index for B,D
    const int kh   = (lane < 16) ? 0 : 2; // K base held by this lane (vgpr0=kh, vgpr1=kh+1)

    // ---- load row with edge padding into sA[0..4095] ----
    const float* xr = x + (size_t)row * M_IN;
    for (int i = lane; i < LPAD; i += 32) {
        int s = i - P0;
        s = s < 0 ? 0 : (s > M_IN - 1 ? M_IN - 1 : s);
        sA[i] = xr[s];
    }
    __syncthreads();

    // =================== forward (analysis) ===================
    for (int lvl = 0; lvl < LEVELS; ++lvl) {
        const int   Llen = LPAD >> lvl;
        const int   H    = Llen >> 1;
        const int   mask = Llen - 1;
        const float* src = (lvl & 1) ? sB : sA;
        float*       dst = (lvl & 1) ? sA : sB;
        float*      dstC = sC + (LPAD - (LPAD >> lvl)); // detail offset for this level

        // A matrices: row0 = h_hi taps, row1 = h_lo taps, rows 2..15 = 0
        v2f af0, af1;
        af0.x = (m == 0) ? hhi(screc, lvl, kh)     : (m == 1) ? hlo(scal, lvl, kh)     : 0.f;
        af0.y = (m == 0) ? hhi(screc, lvl, kh + 1) : (m == 1) ? hlo(scal, lvl, kh + 1) : 0.f;
        af1.x = (m == 0) ? hhi(screc, lvl, 4 + kh)     : (m == 1) ? hlo(scal, lvl, 4 + kh)     : 0.f;
        af1.y = (m == 0) ? hhi(screc, lvl, 4 + kh + 1) : (m == 1) ? hlo(scal, lvl, 4 + kh + 1) : 0.f;

        for (int j0 = 0; j0 < H; j0 += 16) {
            const int j    = j0 + m;
            const int base = 2 * j;
            // B[k,n] = src[(2*j_n - (4p + k)) mod L]
            v2f b0, b1;
            b0.x = src[(base - kh)         & mask];
            b0.y = src[(base - kh - 1)     & mask];
            b1.x = src[(base - 4 - kh)     & mask];
            b1.y = src[(base - 4 - kh - 1) & mask];

            v8f c = {};
            c = __builtin_amdgcn_wmma_f32_16x16x4_f32(false, af0, false, b0, (short)0, c, false, false);
            c = __builtin_amdgcn_wmma_f32_16x16x4_f32(false, af1, false, b1, (short)0, c, false, false);

            if (lane < 16) {
                dstC[j] = c[0]; // D[0, n] = detail[j]
                dst[j]  = c[1]; // D[1, n] = approx'[j]
            }
        }
        __syncthreads();
    }

    // final approx (length 16) sits in sA after level 7 -> tail of coeffs
    if (lane < 16) sC[LPAD - 16 + lane] = sA[lane];
    __syncthreads();

    // ---- stream coeffs to global (contiguous, coalesced) ----
    float* coeffs = out + (size_t)ROWS * M_IN + (size_t)row * LPAD;
    for (int i = lane; i < LPAD; i += 32) coeffs[i] = sC[i];

    // =================== inverse (synthesis) ===================
    for (int lvl = LEVELS - 1; lvl >= 0; --lvl) {
        const int    H      = LPAD >> (lvl + 1);
        const int    Llen   = LPAD >> lvl;
        const int    hmask  = H - 1;
        const int    omask  = Llen - 1;
        const float* srcApx = (lvl & 1) ? sA : sB;
        float*       dst    = (lvl & 1) ? sB : sA;
        const float* d      = sC + (LPAD - (LPAD >> lvl));

        // A: row0 = even taps (even outputs), row1 = odd taps (odd outputs)
        v2f aid, aia;
        aid.x = (m == 0) ? hhi(screc, lvl, 2 * kh)     : (m == 1) ? hhi(screc, lvl, 2 * kh + 1) : 0.f;
        aid.y = (m == 0) ? hhi(screc, lvl, 2 * kh + 2) : (m == 1) ? hhi(screc, lvl, 2 * kh + 3) : 0.f;
        aia.x = (m == 0) ? hlo(scal, lvl, 2 * kh)      : (m == 1) ? hlo(scal, lvl, 2 * kh + 1)  : 0.f;
        aia.y = (m == 0) ? hlo(scal, lvl, 2 * kh + 2)  : (m == 1) ? hlo(scal, lvl, 2 * kh + 3)  : 0.f;

        for (int t0 = 0; t0 < H; t0 += 16) {
            // B[k,n] = coeff[(t0 + n + k) mod H]
            v2f bd, ba;
            bd.x = d[(t0 + m + kh)         & hmask];
            bd.y = d[(t0 + m + kh + 1)     & hmask];
            ba.x = srcApx[(t0 + m + kh)     & hmask];
            ba.y = srcApx[(t0 + m + kh + 1) & hmask];

            v8f c = {};
            c = __builtin_amdgcn_wmma_f32_16x16x4_f32(false, aid, false, bd, (short)0, c, false, false);
            c = __builtin_amdgcn_wmma_f32_16x16x4_f32(false, aia, false, ba, (short)0, c, false, false);

            if (lane < 16) {
                const int t = t0 + m;
                dst[(2 * t)     & omask] = c[0]; // even output at t
                dst[(2 * t - 1) & omask] = c[1]; // odd output at t-1 (wraps at t=0)
            }
        }
        __syncthreads();
    }

    // ---- recon = final approx (in sA, len 4096) cropped to [548, 3548) ----
    float* rec = out + (size_t)row * M_IN;
    for (int i = lane; i < M_IN; i += 32) rec[i] = sA[P0 + i];
}

extern "C" void kernel_launch(void* const* d_in, const int* in_sizes, int n_in,
                              void* d_out, int out_size, void* d_ws, size_t ws_size,
                              hipStream_t stream) {
    (void)in_sizes; (void)n_in; (void)out_size; (void)d_ws; (void)ws_size;
    const float* x     = (const float*)d_in[0];
    const float* scal  = (const float*)d_in[1];
    const float* screc = (const float*)d_in[2];
    float*       out   = (float*)d_out;
    despawn2d_wavelet_kernel<<<dim3(ROWS), dim3(32), 0, stream>>>(x, scal, screc, out);
}